// SqueezeExcitation_13735305412823
// MI455X (gfx1250) — compile-verified
//
#include <hip/hip_runtime.h>

typedef float v2f __attribute__((ext_vector_type(2)));
typedef float v8f __attribute__((ext_vector_type(8)));

#define SE_B   16
#define SE_HW  65536      // 256*256
#define SE_C   256
#define SE_CSQ 64

// ---------------------------------------------------------------- zero sums
__global__ void se_zero(float* __restrict__ sums) {
    sums[blockIdx.x * 256 + threadIdx.x] = 0.0f;
}

// ------------------------------------------------- pass 1: sum over H*W (NHWC)
// grid = (32, 16), block = 256. Each block: batch b = blockIdx.y, a chunk of
// 2048 spatial positions. Thread layout: c4 = tid&63 (channel float4 group),
// psub = tid>>6 (position subgroup of 4). float4 loads are contiguous 1KB per
// position across the 64 c4 lanes -> fully coalesced B128.
__global__ void __launch_bounds__(256) se_reduce(const float* __restrict__ x,
                                                 float* __restrict__ sums) {
    const int b    = blockIdx.y;
    const int tid  = threadIdx.x;
    const int c4   = tid & 63;
    const int psub = tid >> 6;
    const int CHUNK = 2048;

    const long base = ((long)b * SE_HW + (long)blockIdx.x * CHUNK) * SE_C;
    const float4* __restrict__ xv = (const float4*)x;

    float4 acc = make_float4(0.f, 0.f, 0.f, 0.f);
    for (int p = psub; p < CHUNK; p += 4) {
        float4 v = xv[(base + (long)p * SE_C) / 4 + c4];
        acc.x += v.x; acc.y += v.y; acc.z += v.z; acc.w += v.w;
    }

    __shared__ float4 lds[256];
    lds[tid] = acc;
    __syncthreads();

    if (psub == 0) {
        float4 t0 = lds[c4], t1 = lds[c4 + 64], t2 = lds[c4 + 128], t3 = lds[c4 + 192];
        float4 t;
        t.x = t0.x + t1.x + t2.x + t3.x;
        t.y = t0.y + t1.y + t2.y + t3.y;
        t.z = t0.z + t1.z + t2.z + t3.z;
        t.w = t0.w + t1.w + t2.w + t3.w;
        float* s = sums + b * SE_C + c4 * 4;
        atomicAdd(s + 0, t.x);
        atomicAdd(s + 1, t.y);
        atomicAdd(s + 2, t.z);
        atomicAdd(s + 3, t.w);
    }
}

// ------------------------------------- pass 2: gate math with f32 WMMA, 1 wave
// D = A x B + C with V_WMMA_F32_16X16X4_F32 (wave32, EXEC all ones).
// A 16x4 layout : lane m = lane&15 ; VGPR0/1 hold K = k0 + (lane>>4)*2 (+0,+1)
// B 4x16 layout : lane n = lane&15 ; same K selection
// C/D 16x16     : element v -> row v + 8*(lane>>4), col lane&15
__global__ void __launch_bounds__(32) se_gate(const float* __restrict__ sums,
                                              const float* __restrict__ w0,
                                              const float* __restrict__ w1,
                                              const float* __restrict__ w_mult,
                                              float* __restrict__ gate) {
    const int lane = threadIdx.x;
    const int col  = lane & 15;          // A-row m and B/C/D-col n
    const int hi   = lane >> 4;          // lane half: K pair / row-half select

    __shared__ float ymean[SE_B * SE_C];   // 16 KB: pooled means [16][256]
    __shared__ float hid[SE_B * SE_CSQ];   //  4 KB: hidden       [16][64]

    const float inv_hw = 1.0f / (float)SE_HW;
    for (int i = lane; i < SE_B * SE_C; i += 32)
        ymean[i] = sums[i] * inv_hw;
    __syncthreads();

    // ---- GEMM1: [16x256] @ w0[256x64] -> hid, + leaky_relu(0.1)
    for (int nt = 0; nt < SE_CSQ / 16; ++nt) {
        v8f acc = {};
        const int ng = nt * 16 + col;
        for (int k0 = 0; k0 < SE_C; k0 += 4) {
            const int kb = k0 + hi * 2;
            v2f a, bm;
            a.x  = ymean[col * SE_C + kb];
            a.y  = ymean[col * SE_C + kb + 1];
            bm.x = w0[kb * SE_CSQ + ng];
            bm.y = w0[(kb + 1) * SE_CSQ + ng];
            acc = __builtin_amdgcn_wmma_f32_16x16x4_f32(
                false, a, false, bm, (short)0, acc, false, false);
        }
#pragma unroll
        for (int v = 0; v < 8; ++v) {
            float h = acc[v];
            h = (h >= 0.0f) ? h : 0.1f * h;
            hid[(v + hi * 8) * SE_CSQ + ng] = h;
        }
    }
    __syncthreads();

    // ---- GEMM2: hid[16x64] @ w1[64x256], then relu(1+w_mult)*o, sigmoid
    for (int nt = 0; nt < SE_C / 16; ++nt) {
        v8f acc = {};
        const int ng = nt * 16 + col;
        for (int k0 = 0; k0 < SE_CSQ; k0 += 4) {
            const int kb = k0 + hi * 2;
            v2f a, bm;
            a.x  = hid[col * SE_CSQ + kb];
            a.y  = hid[col * SE_CSQ + kb + 1];
            bm.x = w1[kb * SE_C + ng];
            bm.y = w1[(kb + 1) * SE_C + ng];
            acc = __builtin_amdgcn_wmma_f32_16x16x4_f32(
                false, a, false, bm, (short)0, acc, false, false);
        }
        float s = 1.0f + w_mult[ng];
        s = (s > 0.0f) ? s : 0.0f;
#pragma unroll
        for (int v = 0; v < 8; ++v) {
            float o = acc[v] * s;
            float g = 1.0f / (1.0f + __expf(-o));
            gate[(v + hi * 8) * SE_C + ng] = g;
        }
    }
}

// --------------------------------- pass 3: out = x * gate (broadcast over HW)
// grid = (1024, 16), block = 256. Gate row for batch b cached in LDS.
__global__ void __launch_bounds__(256) se_apply(const float* __restrict__ x,
                                                const float* __restrict__ gate,
                                                float* __restrict__ out) {
    const int b = blockIdx.y;
    __shared__ float4 g4[SE_C / 4];
    if (threadIdx.x < SE_C / 4)
        g4[threadIdx.x] = ((const float4*)(gate + b * SE_C))[threadIdx.x];
    __syncthreads();

    const long nvec   = (long)SE_HW * SE_C / 4;   // 4,194,304 float4 per batch
    const long base   = (long)b * nvec;
    const long stride = (long)gridDim.x * blockDim.x;
    const float4* __restrict__ xv = (const float4*)x;
    float4* __restrict__ ov       = (float4*)out;

    for (long i = (long)blockIdx.x * blockDim.x + threadIdx.x; i < nvec; i += stride) {
        float4 xq = xv[base + i];
        float4 gq = g4[i & (SE_C / 4 - 1)];
        float4 r;
        r.x = xq.x * gq.x;
        r.y = xq.y * gq.y;
        r.z = xq.z * gq.z;
        r.w = xq.w * gq.w;
        ov[base + i] = r;
    }
}

extern "C" void kernel_launch(void* const* d_in, const int* in_sizes, int n_in,
                              void* d_out, int out_size, void* d_ws, size_t ws_size,
                              hipStream_t stream) {
    const float* x      = (const float*)d_in[0];   // [16,256,256,256]
    const float* w0     = (const float*)d_in[1];   // [256,64]
    const float* w1     = (const float*)d_in[2];   // [64,256]
    const float* w_mult = (const float*)d_in[3];   // [256]
    float* out = (float*)d_out;

    float* sums = (float*)d_ws;                    // 4096 f32
    float* gate = sums + SE_B * SE_C;              // 4096 f32

    se_zero<<<dim3(SE_B), 256, 0, stream>>>(sums);
    se_reduce<<<dim3(32, SE_B), 256, 0, stream>>>(x, sums);
    se_gate<<<dim3(1), 32, 0, stream>>>(sums, w0, w1, w_mult, gate);
    se_apply<<<dim3(1024, SE_B), 256, 0, stream>>>(x, gate, out);
}